// ReachabilityTimesXModel_83408264888624
// MI455X (gfx1250) — compile-verified
//
#include <hip/hip_runtime.h>

// ---------------------------------------------------------------------------
// ReachabilityTimesX on MI455X (gfx1250, wave32)
//   out[k, i, f] = sum_j coeffs[k, j] * M_j[i, f]
//   M_0 = x;  M_{j+1} = invdeg .* scatter(gather(M_j))  (alternating F/B)
//
// Workspace layout (floats):
//   [0, NP)                      accum   (scatter target, zeroed each step)
//   [NP, NP+N)                   deg
//   [NP+N, NP+2N)                invdeg
//   [NP+2N, NP+2N + 10*NP)       monomials M_1..M_10
// ---------------------------------------------------------------------------

#define N_NODES   100000LL
#define FEAT      128LL
#define NP        (N_NODES * FEAT)     // 12,800,000 elements per monomial
#define NEDGE     3200000LL
#define KOUT      8
#define ORDER     10                   // coeffs has ORDER+1 = 11 columns

typedef float v2f __attribute__((ext_vector_type(2)));
typedef float v8f __attribute__((ext_vector_type(8)));

// Hardware no-return fp32 atomic add (tracked on STOREcnt; S_ENDPGM's implicit
// wait-idle drains it at kernel end). Inline asm guarantees the L2 near-atomic
// path instead of a possible CAS-loop expansion of atomicrmw fadd.
__device__ __forceinline__ void atomicAddF32(float* p, float v) {
  asm volatile("global_atomic_add_f32 %0, %1, off" : : "v"(p), "v"(v) : "memory");
}

// ---- zero a float region ---------------------------------------------------
__global__ void zero_kernel(float* __restrict__ p, long long n) {
  long long i = (long long)blockIdx.x * blockDim.x + threadIdx.x;
  long long stride = (long long)gridDim.x * blockDim.x;
  for (; i < n; i += stride) p[i] = 0.0f;
}

// ---- in-degree: deg[dst[e]] += 1 ------------------------------------------
__global__ void degree_kernel(const long long* __restrict__ dst,
                              float* __restrict__ deg) {
  long long e = (long long)blockIdx.x * blockDim.x + threadIdx.x;
  long long stride = (long long)gridDim.x * blockDim.x;
  for (; e < NEDGE; e += stride)
    atomicAddF32(&deg[dst[e]], 1.0f);
}

// ---- invdeg = 1 / max(deg, 1) ---------------------------------------------
__global__ void invdeg_kernel(const float* __restrict__ deg,
                              float* __restrict__ invdeg) {
  long long i = (long long)blockIdx.x * blockDim.x + threadIdx.x;
  if (i < N_NODES) invdeg[i] = 1.0f / fmaxf(deg[i], 1.0f);
}

// ---- edge-parallel SPMM scatter: accum[s,:] += M[g,:] ----------------------
// One wave per edge. Edge indices are made wave-uniform via readfirstlane so
// they become 2 scalar loads (s_load_b64) instead of 32 redundant lane loads.
// Feature gather: uniform base + lane*16B -> one coalesced global_load_b128
// per lane (monomial array = 51.2MB, L2-resident). Scatter: 4 hw fp32 atomics.
__global__ void spmm_scatter(const float* __restrict__ monoIn,
                             const long long* __restrict__ gidx,
                             const long long* __restrict__ sidx,
                             float* __restrict__ accum) {
  const int lane = threadIdx.x & 31;
  const long long wave  = (long long)blockIdx.x * (blockDim.x >> 5) + (threadIdx.x >> 5);
  const long long nWave = (long long)gridDim.x * (blockDim.x >> 5);
  for (long long e = wave; e < NEDGE; e += nWave) {
    const int eu = __builtin_amdgcn_readfirstlane((int)e);   // wave-uniform
    const long long g = gidx[eu];                            // s_load_b64
    const long long s = sidx[eu];                            // s_load_b64
    const float4 v = *(const float4*)(monoIn + g * FEAT + lane * 4);
    float* dstp = accum + s * FEAT + lane * 4;
    atomicAddF32(dstp + 0, v.x);
    atomicAddF32(dstp + 1, v.y);
    atomicAddF32(dstp + 2, v.z);
    atomicAddF32(dstp + 3, v.w);
  }
}

// ---- scale by invdeg into monomial buffer, and reset accum to zero ---------
__global__ void scale_reset(float* __restrict__ accum,
                            const float* __restrict__ invdeg,
                            float* __restrict__ monoOut) {
  long long i = (long long)blockIdx.x * blockDim.x + threadIdx.x; // float4 idx
  long long stride = (long long)gridDim.x * blockDim.x;
  const long long n4 = NP / 4;
  for (; i < n4; i += stride) {
    const long long node = i >> 5;               // (i*4) / 128
    const float s = invdeg[node];
    float4 v = *(float4*)(accum + i * 4);
    v.x *= s; v.y *= s; v.z *= s; v.w *= s;
    *(float4*)(monoOut + i * 4) = v;
    *(float4*)(accum + i * 4) = make_float4(0.f, 0.f, 0.f, 0.f);
  }
}

// ---- per-lane fetch of monomial j at flat position p -----------------------
__device__ __forceinline__ float loadM(const float* __restrict__ x,
                                       const float* __restrict__ mons,
                                       int j, long long pos) {
  if (j == 0)     return x[pos];
  if (j > ORDER)  return 0.0f;                   // K-padding column
  return mons[(long long)(j - 1) * NP + pos];
}

// ---- final combine with V_WMMA_F32_16X16X4_F32 -----------------------------
// A (16x4): coeffs padded to 16 rows x 12 cols, three K=4 chunks.
// B (4x16): 16 consecutive flat positions, rows = monomial index j.
// D (16x16 f32, v8f): VGPR r, lanes 0-15 -> row M=r  => the 8 real outputs.
__global__ void combine_wmma(const float* __restrict__ x,
                             const float* __restrict__ coeffs,
                             const float* __restrict__ mons,
                             float* __restrict__ out) {
  const int lane = threadIdx.x & 31;
  const int row  = lane & 15;
  const bool hi  = lane >= 16;

  // A fragments (ISA layout: v.x -> K = 4c + (hi?2:0), v.y -> K = 4c + (hi?3:1))
  v2f a[3];
#pragma unroll
  for (int c = 0; c < 3; ++c) {
    const int j0 = 4 * c + (hi ? 2 : 0);
    const int j1 = 4 * c + (hi ? 3 : 1);
    a[c].x = (row < KOUT && j0 <= ORDER) ? coeffs[row * (ORDER + 1) + j0] : 0.f;
    a[c].y = (row < KOUT && j1 <= ORDER) ? coeffs[row * (ORDER + 1) + j1] : 0.f;
  }

  const long long nTiles = NP / 16;  // 800,000 tiles of 16 positions
  const long long waveId = (long long)blockIdx.x * (blockDim.x >> 5) + (threadIdx.x >> 5);
  const long long nWaves = (long long)gridDim.x * (blockDim.x >> 5);

  for (long long t = waveId; t < nTiles; t += nWaves) {
    const long long pos = t * 16 + row;
    v8f acc = {};
#pragma unroll
    for (int c = 0; c < 3; ++c) {
      const int j0 = 4 * c + (hi ? 2 : 0);
      const int j1 = 4 * c + (hi ? 3 : 1);
      v2f b;
      b.x = loadM(x, mons, j0, pos);
      b.y = loadM(x, mons, j1, pos);
      // (neg_a, A, neg_b, B, c_mod, C, reuse_a, reuse_b)
      acc = __builtin_amdgcn_wmma_f32_16x16x4_f32(
          false, a[c], false, b, (short)0, acc, false, false);
    }
    if (!hi) {  // rows 0..7 of D live in lanes 0-15 of VGPRs 0..7
#pragma unroll
      for (int r = 0; r < KOUT; ++r)
        out[(long long)r * NP + t * 16 + row] = acc[r];
    }
  }
}

// ---------------------------------------------------------------------------
extern "C" void kernel_launch(void* const* d_in, const int* in_sizes, int n_in,
                              void* d_out, int out_size, void* d_ws, size_t ws_size,
                              hipStream_t stream) {
  const float*     x      = (const float*)d_in[0];       // [N, 128] f32
  const float*     coeffs = (const float*)d_in[1];       // [8, 11]  f32
  const long long* edge   = (const long long*)d_in[2];   // [2, E]   i64
  const long long* src    = edge;                        // edge_index[0]
  const long long* dst    = edge + NEDGE;                // edge_index[1]
  float*           out    = (float*)d_out;               // [8, N, 128] f32

  float* ws     = (float*)d_ws;
  float* accum  = ws;                       // NP
  float* deg    = ws + NP;                  // N
  float* invd   = deg + N_NODES;            // N
  float* mons   = invd + N_NODES;           // 10 * NP  (M_1 .. M_10)

  // 1) zero accum + deg (contiguous NP + N floats)
  zero_kernel<<<4096, 256, 0, stream>>>(ws, NP + N_NODES);

  // 2) in-degree, then inverse
  degree_kernel<<<4096, 256, 0, stream>>>(dst, deg);
  invdeg_kernel<<<(int)((N_NODES + 255) / 256), 256, 0, stream>>>(deg, invd);

  // 3) ten alternating SPMM steps, materializing M_1..M_10
  const float* cur = x;
  for (int k = 0; k < ORDER; ++k) {
    const long long* g = (k % 2 == 0) ? src : dst;   // 'F' then 'B'
    const long long* s = (k % 2 == 0) ? dst : src;
    spmm_scatter<<<8192, 256, 0, stream>>>(cur, g, s, accum);
    float* mo = mons + (long long)k * NP;
    scale_reset<<<4096, 256, 0, stream>>>(accum, invd, mo);
    cur = mo;
  }

  // 4) fold the 8x11 coefficient matrix over {x, M_1..M_10} with f32 WMMA
  combine_wmma<<<2048, 256, 0, stream>>>(x, coeffs, mons, out);
}